// ReservoirLinearRNN_Block_71897752535272
// MI455X (gfx1250) — compile-verified
//
#include <hip/hip_runtime.h>

typedef __attribute__((ext_vector_type(2))) float v2f;
typedef __attribute__((ext_vector_type(8))) float v8f;

#define HID    1024
#define SEQ    2048
#define BATCH  8
#define CHUNK  32
#define NCHUNK 64          // SEQ / CHUNK
#define SCAN_M 512         // NCHUNK * BATCH
#define SCAN_WGS 64
#define SLICE  16          // HID / SCAN_WGS

#if defined(__has_builtin)
#if __has_builtin(__builtin_amdgcn_tensor_load_to_lds) && \
    __has_builtin(__builtin_amdgcn_s_wait_tensorcnt)
#define HAVE_TDM 1
#endif
#endif

#ifdef HAVE_TDM
typedef unsigned int tdm_v4u __attribute__((ext_vector_type(4)));
typedef int          tdm_v4i __attribute__((ext_vector_type(4)));
typedef int          tdm_v8i __attribute__((ext_vector_type(8)));
#endif

// ---------------- device-coherent state access (bypass stale L0) -----------
__device__ __forceinline__ float2 ld_state2(const float* p) {
  unsigned long long u = __hip_atomic_load((const unsigned long long*)p,
                                           __ATOMIC_RELAXED, __HIP_MEMORY_SCOPE_AGENT);
  return __builtin_bit_cast(float2, u);
}
__device__ __forceinline__ void st_state(float* p, float v) {
  __hip_atomic_store((unsigned int*)p, __builtin_bit_cast(unsigned int, v),
                     __ATOMIC_RELAXED, __HIP_MEMORY_SCOPE_AGENT);
}

// ---------------- device-wide generation barrier (all WGs resident) --------
__device__ __forceinline__ void grid_barrier(int* cnt, int* gen, int nwg) {
  __syncthreads();
  if (threadIdx.x == 0) {
    __threadfence();
    int g = __hip_atomic_load(gen, __ATOMIC_RELAXED, __HIP_MEMORY_SCOPE_AGENT);
    int p = __hip_atomic_fetch_add(cnt, 1, __ATOMIC_ACQ_REL, __HIP_MEMORY_SCOPE_AGENT);
    if (p == nwg - 1) {
      __hip_atomic_store(cnt, 0, __ATOMIC_RELAXED, __HIP_MEMORY_SCOPE_AGENT);
      __hip_atomic_fetch_add(gen, 1, __ATOMIC_ACQ_REL, __HIP_MEMORY_SCOPE_AGENT);
    } else {
      while (__hip_atomic_load(gen, __ATOMIC_ACQUIRE, __HIP_MEMORY_SCOPE_AGENT) == g)
        __builtin_amdgcn_s_sleep(2);
    }
  }
  __syncthreads();
}

// ===========================================================================
// Generic f32 WMMA GEMM:  C[M,N] = act(A[M,K](lda) @ B[K,N] + bias)
// Block tile 128x128, K-chunk 32. 8 waves -> 4(M) x 2(N), 32x64 per wave
// (2x4 v_wmma_f32_16x16x4_f32 tiles -> A-frag reused 4x per load).
// N,K multiples of 128/32; M guarded.
// ===========================================================================
#define GBM 128
#define GBN 128
#define GBK 32
#define GLDK (GBK + 2)

__global__ __launch_bounds__(256) void gemm_f32_wmma(
    const float* __restrict__ A, long lda,
    const float* __restrict__ B,
    const float* __restrict__ bias,
    float* __restrict__ C,
    int M, int N, int K, int relu)
{
  __shared__ float As[GBM * GLDK];   // [row][k]  (padded)
  __shared__ float Bs[GBN * GLDK];   // [col][k]  (transposed, padded)

  const int tid  = threadIdx.x;
  const int lane = tid & 31;
  const int wid  = tid >> 5;
  const int wm   = wid & 3;          // 0..3  (M direction, 32 rows each)
  const int wn   = wid >> 2;         // 0..1  (N direction, 64 cols each)
  const int m0   = blockIdx.y * GBM;
  const int n0   = blockIdx.x * GBN;

  v8f acc[2][4] = {};

  for (int kk = 0; kk < K; kk += GBK) {
    __syncthreads();
    // ---- stage A tile: 128 rows x 32 cols, float4 per thread x4 ----
    for (int q = tid; q < (GBM * GBK) / 4; q += 256) {
      int row = q >> 3;
      int c4  = (q & 7) << 2;
      float4 v = make_float4(0.f, 0.f, 0.f, 0.f);
      int gr = m0 + row;
      if (gr < M) v = *(const float4*)(A + (long)gr * lda + kk + c4);
      float* dst = &As[row * GLDK + c4];
      dst[0] = v.x; dst[1] = v.y; dst[2] = v.z; dst[3] = v.w;
    }
    // ---- stage B tile transposed: [col][k] ----
    for (int q = tid; q < GBN * GBK; q += 256) {
      int k = q >> 7;
      int n = q & 127;
      Bs[n * GLDK + k] = B[(long)(kk + k) * N + n0 + n];
    }
    __syncthreads();

    const int koff = (lane >> 4) << 1;   // 0 or 2
    const int lidx = lane & 15;
#pragma unroll
    for (int k4 = 0; k4 < GBK; k4 += 4) {
      v2f a[2], b[4];
#pragma unroll
      for (int i = 0; i < 2; ++i) {
        const float* p = &As[(wm * 32 + i * 16 + lidx) * GLDK + k4 + koff];
        a[i][0] = p[0]; a[i][1] = p[1];
      }
#pragma unroll
      for (int j = 0; j < 4; ++j) {
        const float* p = &Bs[(wn * 64 + j * 16 + lidx) * GLDK + k4 + koff];
        b[j][0] = p[0]; b[j][1] = p[1];
      }
#pragma unroll
      for (int i = 0; i < 2; ++i)
#pragma unroll
        for (int j = 0; j < 4; ++j)
          acc[i][j] = __builtin_amdgcn_wmma_f32_16x16x4_f32(
              false, a[i], false, b[j], (short)0, acc[i][j], false, false);
    }
  }

  // ---- epilogue: C layout = VGPR e -> rows e / e+8, lanes -> 16 cols ----
  const int colL  = lane & 15;
  const int rhalf = (lane >> 4) << 3;
#pragma unroll
  for (int i = 0; i < 2; ++i)
#pragma unroll
    for (int j = 0; j < 4; ++j) {
      int col = n0 + wn * 64 + j * 16 + colL;
      float bv = bias ? bias[col] : 0.f;
#pragma unroll
      for (int e = 0; e < 8; ++e) {
        int row = m0 + wm * 32 + i * 16 + rhalf + e;
        if (row < M) {
          float v = acc[i][j][e] + bv;
          if (relu) v = fmaxf(v, 0.f);
          C[(long)row * N + col] = v;
        }
      }
    }
}

// ===========================================================================
// Persistent chunked-scan kernel.  64 WGs; WG owns a 16-col slice of Amat
// staged into LDS by the Tensor Data Mover (2D tile: 16 x 1024, stride 1024).
// LDS layout [k][16] (packed TDM order) -- bank-conflict-free for B-frags.
// State (Mrows x 1024) double-buffered in global (agent-scope accesses).
// Per step:  S_next = S_cur @ Amat + B_t  ; device-wide barrier between steps.
//   mode 0 (stage1): Bseq=Bu (chunked), init 0,      final -> outFinal (PL)
//   mode 1 (carry):  Bseq=PL,           init 0,      every -> outAll (Carry[t+1])
//   mode 2 (stage3): Bseq=Bu (chunked), init=Carry,  every -> outAll (H_pre)
// ===========================================================================
__global__ __launch_bounds__(256) void scan_wmma(
    const float* __restrict__ Amat,
    const float* __restrict__ Bseq,
    const float* __restrict__ initS,
    float* S0, float* S1,
    float* __restrict__ outFinal,
    float* __restrict__ outAll,
    int Mrows, int steps, int mode,
    int* bar_cnt, int* bar_gen)
{
  __shared__ float Alds[HID * 16];   // 64 KB: element (k, c) at k*16 + c
  const int n0   = blockIdx.x * SLICE;
  const int tid  = threadIdx.x;
  const int lane = tid & 31;
  const int wid  = tid >> 5;

#ifdef HAVE_TDM
  // ---- TDM: async DMA of the strided A-slice into LDS ----
  if (tid == 0) {
    unsigned long long ga = (unsigned long long)(uintptr_t)(Amat + n0);
    unsigned lds_off = (unsigned)(uintptr_t)&Alds[0];
    tdm_v4u g0;
    g0[0] = 1u;                                   // count=1 (valid user D#)
    g0[1] = lds_off;                              // lds_addr
    g0[2] = (unsigned)(ga & 0xffffffffu);         // global_addr[31:0]
    g0[3] = (unsigned)((ga >> 32) & 0x01ffffffu)  // global_addr[56:32]
            | (2u << 30);                         // type = 2 ("image")
    tdm_v8i g1;
    g1[0] = (int)(2u << 16);                      // data_size = 4B, no mask
    g1[1] = (int)(16u << 16);                     // tensor_dim0 = 16
    g1[2] = (int)(1024u << 16);                   // tensor_dim1 = 1024
    g1[3] = (int)(16u << 16);                     // tile_dim0 = 16
    g1[4] = (int)1024;                            // tile_dim1 = 1024
    g1[5] = (int)1024;                            // tensor_dim0_stride = 1024
    g1[6] = 0;
    g1[7] = 0;
    tdm_v4i gz4 = {};
    tdm_v8i gz8 = {};
    __builtin_amdgcn_tensor_load_to_lds(g0, g1, gz4, gz4, gz8, 0);
    __builtin_amdgcn_s_wait_tensorcnt(0);
  }
#else
  for (int idx = tid; idx < HID * SLICE; idx += 256) {
    int k = idx >> 4, c = idx & 15;
    Alds[idx] = Amat[(long)k * HID + n0 + c];
  }
#endif

  // init this WG's state slice
  for (int idx = tid; idx < Mrows * SLICE; idx += 256) {
    int r = idx / SLICE, c = idx - r * SLICE;
    float v = initS ? initS[(long)r * HID + n0 + c] : 0.0f;
    st_state(&S0[(long)r * HID + n0 + c], v);
  }
  grid_barrier(bar_cnt, bar_gen, gridDim.x);

  float* Scur = S0;
  float* Snxt = S1;
  const int Mt    = (Mrows + 15) >> 4;
  const int colL  = lane & 15;
  const int koff  = (lane >> 4) << 1;     // 0 / 2
  const int rhalf = (lane >> 4) << 3;     // 0 / 8
  const float* bcol = &Alds[colL];        // column colL, stride 16

  for (int t = 0; t < steps; ++t) {
    for (int mt = wid; mt < Mt; mt += 8) {
      const int mbase = mt << 4;
      const int arow  = mbase + colL;
      const bool rowok = arow < Mrows;
      const float* srow = Scur + (long)arow * HID + koff;
      v8f acc = {};
#pragma unroll 8
      for (int k = 0; k < HID; k += 4) {
        float2 af = rowok ? ld_state2(srow + k) : make_float2(0.f, 0.f);
        const int kb = k + koff;
        v2f a; a[0] = af.x;           a[1] = af.y;
        v2f b; b[0] = bcol[kb << 4];  b[1] = bcol[(kb + 1) << 4];
        acc = __builtin_amdgcn_wmma_f32_16x16x4_f32(
            false, a, false, b, (short)0, acc, false, false);
      }
      const int col = n0 + colL;
#pragma unroll
      for (int e = 0; e < 8; ++e) {
        int r = mbase + rhalf + e;
        if (r < Mrows) {
          float bu;
          long orow = 0;
          if (mode == 1) {
            bu = Bseq[((long)t * BATCH + r) * HID + col];
          } else {
            int bb = r & 7, cc = r >> 3;
            orow = (long)bb * SEQ + (long)cc * CHUNK + t;
            bu = Bseq[orow * HID + col];
          }
          float v = acc[e] + bu;
          st_state(&Snxt[(long)r * HID + col], v);
          if (mode == 0) { if (t == steps - 1) outFinal[(long)r * HID + col] = v; }
          else if (mode == 1) { outAll[((long)(t + 1) * BATCH + r) * HID + col] = v; }
          else { outAll[orow * HID + col] = v; }
        }
      }
    }
    grid_barrier(bar_cnt, bar_gen, gridDim.x);
    float* tp = Scur; Scur = Snxt; Snxt = tp;
  }
}

// ---------------- fused z = Hpre[:,S-1,:] + residual ; LayerNorm -----------
__global__ __launch_bounds__(256) void ln_fuse(
    const float* __restrict__ Hpre, const float* __restrict__ resl,
    const float* __restrict__ gamma, const float* __restrict__ beta,
    float* __restrict__ hlast)
{
  __shared__ float red[256];
  int b = blockIdx.x, tid = threadIdx.x;
  const float* hr = Hpre + ((long)b * SEQ + (SEQ - 1)) * HID;
  const float* rr = resl + (long)b * HID;
  float z[4]; float s = 0.f, s2 = 0.f;
#pragma unroll
  for (int i = 0; i < 4; ++i) {
    int n = tid + (i << 8);
    z[i] = hr[n] + rr[n];
    s += z[i]; s2 += z[i] * z[i];
  }
  red[tid] = s; __syncthreads();
  for (int o = 128; o > 0; o >>= 1) { if (tid < o) red[tid] += red[tid + o]; __syncthreads(); }
  float mean = red[0] * (1.0f / HID);
  __syncthreads();
  red[tid] = s2; __syncthreads();
  for (int o = 128; o > 0; o >>= 1) { if (tid < o) red[tid] += red[tid + o]; __syncthreads(); }
  float var = red[0] * (1.0f / HID) - mean * mean;
  float rs = rsqrtf(var + 1e-5f);
#pragma unroll
  for (int i = 0; i < 4; ++i) {
    int n = tid + (i << 8);
    hlast[(long)b * HID + n] = (z[i] - mean) * rs * gamma[n] + beta[n];
  }
}

__global__ void init_misc(int* bar, float* carry0) {
  int t = blockIdx.x * blockDim.x + threadIdx.x;
  if (t < 2) bar[t] = 0;
  if (t < BATCH * HID) carry0[t] = 0.f;
}

// ===========================================================================
extern "C" void kernel_launch(void* const* d_in, const int* in_sizes, int n_in,
                              void* d_out, int out_size, void* d_ws, size_t ws_size,
                              hipStream_t stream) {
  const float* x     = (const float*)d_in[0];   // [8,2048,512]
  const float* W_enc = (const float*)d_in[1];   // [512,1024]
  const float* b_enc = (const float*)d_in[2];   // [1024]
  const float* W_B   = (const float*)d_in[3];   // [1024,1024]
  const float* A     = (const float*)d_in[4];   // [1024,1024]
  const float* W_res = (const float*)d_in[5];   // [512,1024]
  const float* b_res = (const float*)d_in[6];
  const float* gamma = (const float*)d_in[7];
  const float* beta  = (const float*)d_in[8];
  const float* W1    = (const float*)d_in[9];   // [1024,4096]
  const float* b1    = (const float*)d_in[10];
  const float* W2    = (const float*)d_in[11];  // [4096,1024]
  const float* b2    = (const float*)d_in[12];

  float* out  = (float*)d_out;       // [8,1024]
  float* Hpre = out + BATCH * HID;   // [8,2048,1024]

  float* ws    = (float*)d_ws;
  float* Bu    = ws; ws += (size_t)BATCH * SEQ * HID;   // 16M
  float* Wcomb = ws; ws += (size_t)512 * HID;
  float* bcomb = ws; ws += HID;
  float* P0    = ws; ws += (size_t)HID * HID;
  float* P1    = ws; ws += (size_t)HID * HID;
  float* PL    = ws; ws += (size_t)SCAN_M * HID;
  float* Carry = ws; ws += (size_t)SCAN_M * HID;
  float* S0    = ws; ws += (size_t)SCAN_M * HID;
  float* S1    = ws; ws += (size_t)SCAN_M * HID;
  float* resl  = ws; ws += (size_t)BATCH * HID;
  float* hlast = ws; ws += (size_t)BATCH * HID;
  float* mlph  = ws; ws += (size_t)BATCH * 4096;
  int*   bar   = (int*)ws;

  auto gemm = [&](const float* Ap, long lda, const float* Bp, const float* biasp,
                  float* Cp, int M, int N, int K, int relu) {
    dim3 g(N / GBN, (M + GBM - 1) / GBM);
    gemm_f32_wmma<<<g, dim3(256), 0, stream>>>(Ap, lda, Bp, biasp, Cp, M, N, K, relu);
  };

  init_misc<<<32, 256, 0, stream>>>(bar, Carry);

  // Fold encoder: Wcomb = W_enc @ W_B ; bcomb = b_enc @ W_B
  gemm(W_enc, HID, W_B, nullptr, Wcomb, 512, HID, HID, 0);
  gemm(b_enc, HID, W_B, nullptr, bcomb, 1,   HID, HID, 0);
  // Bu = x @ Wcomb + bcomb   (M = B*S = 16384, K = 512)
  gemm(x, 512, Wcomb, bcomb, Bu, BATCH * SEQ, HID, 512, 0);

  // A^32 by repeated squaring: A2->P0, A4->P1, A8->P0, A16->P1, A32->P0
  gemm(A,  HID, A,  nullptr, P0, HID, HID, HID, 0);
  gemm(P0, HID, P0, nullptr, P1, HID, HID, HID, 0);
  gemm(P1, HID, P1, nullptr, P0, HID, HID, HID, 0);
  gemm(P0, HID, P0, nullptr, P1, HID, HID, HID, 0);
  gemm(P1, HID, P1, nullptr, P0, HID, HID, HID, 0);

  // Stage 1: local chunk scans (zero init) -> PL
  scan_wmma<<<SCAN_WGS, 256, 0, stream>>>(A, Bu, nullptr, S0, S1, PL, nullptr,
                                          SCAN_M, CHUNK, 0, bar, bar + 1);
  // Carry: e_t = e_{t-1} @ A^32 + PL[t]  -> Carry[t+1]   (Carry[0] zeroed)
  scan_wmma<<<SCAN_WGS, 256, 0, stream>>>(P0, PL, nullptr, S0, S1, nullptr, Carry,
                                          BATCH, NCHUNK - 1, 1, bar, bar + 1);
  // Stage 3: re-scan chunks from true carries, emitting H_pre
  scan_wmma<<<SCAN_WGS, 256, 0, stream>>>(A, Bu, Carry, S0, S1, nullptr, Hpre,
                                          SCAN_M, CHUNK, 2, bar, bar + 1);

  // residual at t = S-1 only:  resl = x[:,S-1,:] @ W_res + b_res
  gemm(x + (long)(SEQ - 1) * 512, (long)SEQ * 512, W_res, b_res, resl,
       BATCH, HID, 512, 0);
  // LayerNorm(z_last) -> hlast
  ln_fuse<<<BATCH, 256, 0, stream>>>(Hpre, resl, gamma, beta, hlast);
  // MLP head
  gemm(hlast, HID, W1, b1, mlph, BATCH, 4096, HID, 1);
  gemm(mlph, 4096, W2, b2, out, BATCH, HID, 4096, 0);
}